// MA_Layer3_15204184228508
// MI455X (gfx1250) — compile-verified
//
#include <hip/hip_runtime.h>
#include <math.h>

// MI455X / gfx1250, wave32. All GEMMs via V_WMMA_F32_16X16X4_F32 (exact f32).
#define D_MODEL 512
#define D_STATE 16
#define DT_RANK 32
#define D_INNER 512
#define BATCH   4
#define SEQ     2048
#define T_TOK   (BATCH*SEQ)   // 8192 tokens
#define LN_EPS  1e-5f
#define AROW    516           // 512 + 4 pad: 2064B row stride -> conflict-free b64 LDS reads

typedef __attribute__((ext_vector_type(2))) float v2f;
typedef __attribute__((ext_vector_type(8))) float v8f;

__device__ __forceinline__ v8f wmma_f32(v2f a, v2f b, v8f c) {
  // D = A(16x4,f32) * B(4x16,f32) + C(16x16,f32)
  return __builtin_amdgcn_wmma_f32_16x16x4_f32(false, a, false, b, (short)0, c,
                                               false, false);
}

__device__ __forceinline__ float sigmoidf_(float x) { return 1.f / (1.f + __expf(-x)); }
__device__ __forceinline__ float siluf_(float x)    { return x * sigmoidf_(x); }
__device__ __forceinline__ float softplusf_(float x){ return (x > 20.f) ? x : log1pf(__expf(x)); }

// ---------------------------------------------------------------------------
// k0: per-batch folded bias  sb[b][j] = gb1[j] + sum_k x_stats[b][k]*gW1[j][512+k]
// ---------------------------------------------------------------------------
__global__ __launch_bounds__(512) void k0_sb(const float* __restrict__ x_stats,
                                             const float* __restrict__ gW1,
                                             const float* __restrict__ gb1,
                                             float* __restrict__ sb) {
  int b = blockIdx.x;
  int j = threadIdx.x;
  const float* xs = x_stats + b * D_MODEL;
  const float* w  = gW1 + (size_t)j * (2 * D_MODEL) + D_MODEL;
  float acc = gb1[j];
  #pragma unroll 4
  for (int k = 0; k < D_MODEL; ++k) acc += xs[k] * w[k];
  sb[b * D_MODEL + j] = acc;
}

// ---------------------------------------------------------------------------
// k1_gate: WMMA GEMM h = relu(x_signal @ gW1a^T + sb), fused gW2 dot + sigmoid
// block = 256 thr (8 waves), 16 tokens per block; each wave does 4 N-tiles.
// ---------------------------------------------------------------------------
__global__ __launch_bounds__(256) void k1_gate(const float* __restrict__ x_signal,
                                               const float* __restrict__ gW1,
                                               const float* __restrict__ gW2,
                                               const float* __restrict__ gb2,
                                               const float* __restrict__ sb,
                                               float* __restrict__ gbuf) {
  __shared__ float As[16 * AROW];
  __shared__ float gpart[16];
  int t0  = blockIdx.x * 16;
  int b   = t0 / SEQ;
  int tid = threadIdx.x;
  for (int idx = tid; idx < 16 * 128; idx += 256) {
    int m = idx >> 7, k4 = (idx & 127) * 4;
    *(float4*)&As[m * AROW + k4] = *(const float4*)&x_signal[(size_t)(t0 + m) * D_MODEL + k4];
  }
  if (tid < 16) gpart[tid] = 0.f;
  __syncthreads();

  int wave = tid >> 5, lane = tid & 31, half = lane >> 4, r16 = lane & 15;
  for (int r = 0; r < 4; ++r) {
    int j = wave * 64 + r * 16 + r16;               // output column (0..511)
    const float* wrow = gW1 + (size_t)j * (2 * D_MODEL);
    v8f c = {};
    #pragma unroll 4
    for (int k = 0; k < D_MODEL; k += 4) {
      v2f a  = *(const v2f*)&As[r16 * AROW + k + 2 * half];
      v2f bf = *(const v2f*)&wrow[k + 2 * half];
      c = wmma_f32(a, bf, c);
    }
    float g2 = gW2[j];
    float sv = sb[b * D_MODEL + j];
    float acc[8];
    #pragma unroll
    for (int v = 0; v < 8; ++v) {
      float h = c[v] + sv;
      h = h > 0.f ? h : 0.f;                         // relu
      acc[v] = h * g2;
    }
    // reduce over the 16 lanes that share the same M rows (xor masks stay in half)
    #pragma unroll
    for (int mask = 1; mask < 16; mask <<= 1)
      #pragma unroll
      for (int v = 0; v < 8; ++v) acc[v] += __shfl_xor(acc[v], mask, 32);
    if (r16 == 0) {
      #pragma unroll
      for (int v = 0; v < 8; ++v) atomicAdd(&gpart[v + 8 * half], acc[v]);
    }
  }
  __syncthreads();
  if (tid < 16) gbuf[t0 + tid] = sigmoidf_(gpart[tid] + gb2[0]);
}

// ---------------------------------------------------------------------------
// k1_ln: x = x_signal + g*xs ; LayerNorm -> xn.  block per token.
// ---------------------------------------------------------------------------
__global__ __launch_bounds__(128) void k1_ln(const float* __restrict__ x_signal,
                                             const float* __restrict__ x_stats,
                                             const float* __restrict__ gbuf,
                                             const float* __restrict__ ln_g,
                                             const float* __restrict__ ln_b,
                                             float* __restrict__ xn) {
  int t = blockIdx.x, b = t / SEQ, tid = threadIdx.x;
  float g = gbuf[t];
  float xi[4], s1 = 0.f, s2 = 0.f;
  #pragma unroll
  for (int q = 0; q < 4; ++q) {
    int i = tid * 4 + q;
    float x = x_signal[(size_t)t * D_MODEL + i] + g * x_stats[b * D_MODEL + i];
    xi[q] = x; s1 += x; s2 += x * x;
  }
  __shared__ float r1[128], r2[128];
  r1[tid] = s1; r2[tid] = s2; __syncthreads();
  for (int off = 64; off > 0; off >>= 1) {
    if (tid < off) { r1[tid] += r1[tid + off]; r2[tid] += r2[tid + off]; }
    __syncthreads();
  }
  float mu  = r1[0] * (1.f / D_MODEL);
  float var = r2[0] * (1.f / D_MODEL) - mu * mu;
  float rs  = rsqrtf(var + LN_EPS);
  #pragma unroll
  for (int q = 0; q < 4; ++q) {
    int i = tid * 4 + q;
    xn[(size_t)t * D_MODEL + i] = (xi[q] - mu) * rs * ln_g[i] + ln_b[i];
  }
}

// ---------------------------------------------------------------------------
// k2: in_proj  xz = xn @ in_proj_W^T (8192x1024x512), split into u | z.
// ---------------------------------------------------------------------------
__global__ __launch_bounds__(256) void k2_inproj(const float* __restrict__ xn,
                                                 const float* __restrict__ inW,
                                                 float* __restrict__ u,
                                                 float* __restrict__ z) {
  __shared__ float As[16 * AROW];
  int t0 = blockIdx.x * 16, tid = threadIdx.x;
  for (int idx = tid; idx < 16 * 128; idx += 256) {
    int m = idx >> 7, k4 = (idx & 127) * 4;
    *(float4*)&As[m * AROW + k4] = *(const float4*)&xn[(size_t)(t0 + m) * D_MODEL + k4];
  }
  __syncthreads();
  int wave = tid >> 5, lane = tid & 31, half = lane >> 4, r16 = lane & 15;
  int n = blockIdx.y * 128 + wave * 16 + r16;       // 0..1023
  const float* wrow = inW + (size_t)n * D_MODEL;
  v8f c = {};
  #pragma unroll 4
  for (int k = 0; k < D_MODEL; k += 4) {
    v2f a  = *(const v2f*)&As[r16 * AROW + k + 2 * half];
    v2f bf = *(const v2f*)&wrow[k + 2 * half];
    c = wmma_f32(a, bf, c);
  }
  float* dst = (n < D_INNER) ? (u + n) : (z + (n - D_INNER));
  #pragma unroll
  for (int v = 0; v < 8; ++v) {
    int m = t0 + v + 8 * half;
    dst[(size_t)m * D_INNER] = c[v];
  }
}

// ---------------------------------------------------------------------------
// k3_conv: depthwise causal conv (K=2) + SiLU  -> us
// ---------------------------------------------------------------------------
__global__ __launch_bounds__(256) void k3_conv(const float* __restrict__ u,
                                               const float* __restrict__ conv_w,
                                               const float* __restrict__ conv_b,
                                               float* __restrict__ us) {
  int idx = blockIdx.x * 256 + threadIdx.x;         // = t*512 + d
  int t = idx >> 9, d = idx & 511;
  int l = t & (SEQ - 1);
  float up = (l > 0) ? u[(size_t)(t - 1) * D_INNER + d] : 0.f;
  float uc = conv_b[d] + conv_w[d * 2] * up + conv_w[d * 2 + 1] * u[(size_t)t * D_INNER + d];
  us[idx] = siluf_(uc);
}

// ---------------------------------------------------------------------------
// k3_xproj: xdbc = us @ x_proj_W^T (8192x64x512)
// ---------------------------------------------------------------------------
__global__ __launch_bounds__(128) void k3_xproj(const float* __restrict__ us,
                                                const float* __restrict__ xW,
                                                float* __restrict__ xdbc) {
  __shared__ float As[16 * AROW];
  int t0 = blockIdx.x * 16, tid = threadIdx.x;
  for (int idx = tid; idx < 16 * 128; idx += 128) {
    int m = idx >> 7, k4 = (idx & 127) * 4;
    *(float4*)&As[m * AROW + k4] = *(const float4*)&us[(size_t)(t0 + m) * D_INNER + k4];
  }
  __syncthreads();
  int wave = tid >> 5, lane = tid & 31, half = lane >> 4, r16 = lane & 15;
  int n = wave * 16 + r16;                          // 0..63
  const float* wrow = xW + (size_t)n * D_INNER;
  v8f c = {};
  #pragma unroll 4
  for (int k = 0; k < D_INNER; k += 4) {
    v2f a  = *(const v2f*)&As[r16 * AROW + k + 2 * half];
    v2f bf = *(const v2f*)&wrow[k + 2 * half];
    c = wmma_f32(a, bf, c);
  }
  #pragma unroll
  for (int v = 0; v < 8; ++v) {
    int m = t0 + v + 8 * half;
    xdbc[(size_t)m * 64 + n] = c[v];
  }
}

// ---------------------------------------------------------------------------
// k3_dtproj: delta = softplus(dt @ dt_proj_W^T + dt_proj_b)  (K = 32)
// ---------------------------------------------------------------------------
__global__ __launch_bounds__(256) void k3_dtproj(const float* __restrict__ xdbc,
                                                 const float* __restrict__ dtW,
                                                 const float* __restrict__ dtb,
                                                 float* __restrict__ delta) {
  __shared__ float As[16 * 36];                     // pad 36 -> bank rotation
  int t0 = blockIdx.x * 16, tid = threadIdx.x;
  for (int idx = tid; idx < 16 * 32; idx += 256) {
    int m = idx >> 5, k = idx & 31;
    As[m * 36 + k] = xdbc[(size_t)(t0 + m) * 64 + k];
  }
  __syncthreads();
  int wave = tid >> 5, lane = tid & 31, half = lane >> 4, r16 = lane & 15;
  int n = blockIdx.y * 128 + wave * 16 + r16;       // 0..511
  const float* wrow = dtW + (size_t)n * DT_RANK;
  v8f c = {};
  #pragma unroll
  for (int k = 0; k < DT_RANK; k += 4) {
    v2f a  = *(const v2f*)&As[r16 * 36 + k + 2 * half];
    v2f bf = *(const v2f*)&wrow[k + 2 * half];
    c = wmma_f32(a, bf, c);
  }
  float bv = dtb[n];
  #pragma unroll
  for (int v = 0; v < 8; ++v) {
    int m = t0 + v + 8 * half;
    delta[(size_t)m * D_INNER + n] = softplusf_(c[v] + bv);
  }
}

// ---------------------------------------------------------------------------
// k4_scan: selective scan, sequential in L; lane = one channel, 16 states in regs.
// grid = 4 batches * 4 channel-chunks, 128 threads.
// ---------------------------------------------------------------------------
__global__ __launch_bounds__(128) void k4_scan(const float* __restrict__ delta,
                                               const float* __restrict__ us,
                                               const float* __restrict__ xdbc,
                                               const float* __restrict__ A_log,
                                               float* __restrict__ yscan) {
  __shared__ float bc[32];                          // B[16] | C[16] for current t
  int b  = blockIdx.x >> 2;
  int d  = (blockIdx.x & 3) * 128 + threadIdx.x;
  float a[D_STATE], h[D_STATE];
  #pragma unroll
  for (int n = 0; n < D_STATE; ++n) { a[n] = -__expf(A_log[d * D_STATE + n]); h[n] = 0.f; }
  for (int l = 0; l < SEQ; ++l) {
    int t = b * SEQ + l;
    if (threadIdx.x < 32) bc[threadIdx.x] = xdbc[(size_t)t * 64 + 32 + threadIdx.x];
    __syncthreads();
    float dl = delta[(size_t)t * D_INNER + d];
    float uu = us[(size_t)t * D_INNER + d];
    float du = dl * uu;
    float y = 0.f;
    #pragma unroll
    for (int n = 0; n < D_STATE; ++n) {
      h[n] = __expf(dl * a[n]) * h[n] + du * bc[n];
      y += h[n] * bc[16 + n];
    }
    yscan[(size_t)t * D_INNER + d] = y;
    __syncthreads();
  }
}

// ---------------------------------------------------------------------------
// k5: fused  y = (yscan + Dp*us)*silu(z) ; out = y @ out_proj_W^T (8192x512x512)
// ---------------------------------------------------------------------------
__global__ __launch_bounds__(256) void k5_out(const float* __restrict__ yscan,
                                              const float* __restrict__ us,
                                              const float* __restrict__ z,
                                              const float* __restrict__ Dp,
                                              const float* __restrict__ oW,
                                              float* __restrict__ out) {
  __shared__ float As[16 * AROW];
  int t0 = blockIdx.x * 16, tid = threadIdx.x;
  for (int idx = tid; idx < 16 * 128; idx += 256) {
    int m = idx >> 7, k4 = (idx & 127) * 4;
    size_t base = (size_t)(t0 + m) * D_INNER + k4;
    float4 yv = *(const float4*)&yscan[base];
    float4 uv = *(const float4*)&us[base];
    float4 zv = *(const float4*)&z[base];
    float4 dv = *(const float4*)&Dp[k4];
    float4 r;
    r.x = (yv.x + dv.x * uv.x) * siluf_(zv.x);
    r.y = (yv.y + dv.y * uv.y) * siluf_(zv.y);
    r.z = (yv.z + dv.z * uv.z) * siluf_(zv.z);
    r.w = (yv.w + dv.w * uv.w) * siluf_(zv.w);
    *(float4*)&As[m * AROW + k4] = r;
  }
  __syncthreads();
  int wave = tid >> 5, lane = tid & 31, half = lane >> 4, r16 = lane & 15;
  int n = blockIdx.y * 128 + wave * 16 + r16;       // 0..511
  const float* wrow = oW + (size_t)n * D_MODEL;
  v8f c = {};
  #pragma unroll 4
  for (int k = 0; k < D_MODEL; k += 4) {
    v2f a  = *(const v2f*)&As[r16 * AROW + k + 2 * half];
    v2f bf = *(const v2f*)&wrow[k + 2 * half];
    c = wmma_f32(a, bf, c);
  }
  #pragma unroll
  for (int v = 0; v < 8; ++v) {
    int m = t0 + v + 8 * half;
    out[(size_t)m * D_MODEL + n] = c[v];
  }
}

// ---------------------------------------------------------------------------
extern "C" void kernel_launch(void* const* d_in, const int* in_sizes, int n_in,
                              void* d_out, int out_size, void* d_ws, size_t ws_size,
                              hipStream_t stream) {
  const float* x_signal = (const float*)d_in[0];
  const float* x_stats  = (const float*)d_in[1];
  const float* gW1      = (const float*)d_in[2];
  const float* gb1      = (const float*)d_in[3];
  const float* gW2      = (const float*)d_in[4];
  const float* gb2      = (const float*)d_in[5];
  const float* ln_g     = (const float*)d_in[6];
  const float* ln_b     = (const float*)d_in[7];
  const float* inW      = (const float*)d_in[8];
  const float* conv_w   = (const float*)d_in[9];
  const float* conv_b   = (const float*)d_in[10];
  const float* xW       = (const float*)d_in[11];
  const float* dtW      = (const float*)d_in[12];
  const float* dtb      = (const float*)d_in[13];
  const float* A_log    = (const float*)d_in[14];
  const float* Dp       = (const float*)d_in[15];
  const float* oW       = (const float*)d_in[16];
  float* out = (float*)d_out;

  float* ws = (float*)d_ws;
  size_t off = 0;
  float* sb    = ws + off; off += (size_t)BATCH * D_MODEL;
  float* gbuf  = ws + off; off += (size_t)T_TOK;
  float* xn    = ws + off; off += (size_t)T_TOK * D_MODEL;   // reused as yscan
  float* u     = ws + off; off += (size_t)T_TOK * D_INNER;
  float* z     = ws + off; off += (size_t)T_TOK * D_INNER;
  float* us    = ws + off; off += (size_t)T_TOK * D_INNER;
  float* delta = ws + off; off += (size_t)T_TOK * D_INNER;
  float* xdbc  = ws + off; off += (size_t)T_TOK * 64;
  float* yscan = xn;   // xn dead after k2; reuse for scan output

  k0_sb    <<<BATCH, D_MODEL, 0, stream>>>(x_stats, gW1, gb1, sb);
  k1_gate  <<<T_TOK / 16, 256, 0, stream>>>(x_signal, gW1, gW2, gb2, sb, gbuf);
  k1_ln    <<<T_TOK, 128, 0, stream>>>(x_signal, x_stats, gbuf, ln_g, ln_b, xn);
  k2_inproj<<<dim3(T_TOK / 16, 8), 256, 0, stream>>>(xn, inW, u, z);
  k3_conv  <<<(T_TOK * D_INNER) / 256, 256, 0, stream>>>(u, conv_w, conv_b, us);
  k3_xproj <<<T_TOK / 16, 128, 0, stream>>>(us, xW, xdbc);
  k3_dtproj<<<dim3(T_TOK / 16, 4), 256, 0, stream>>>(xdbc, dtW, dtb, delta);
  k4_scan  <<<16, 128, 0, stream>>>(delta, us, xdbc, A_log, yscan);
  k5_out   <<<dim3(T_TOK / 16, 4), 256, 0, stream>>>(yscan, us, z, Dp, oW, out);
}